// EdgeLoss_86354612453484
// MI455X (gfx1250) — compile-verified
//
#include <hip/hip_runtime.h>
#include <stdint.h>
#include <math.h>

// ---------------------------------------------------------------------------
// MI455X (gfx1250) implementation. Workload is bandwidth-bound (~0.25 FLOP/B):
//   * dominant traffic = 2 x 260B row gathers per edge (~520 MB) -> wave-per-
//     edge coalesced gathers, double-buffered through LDS with the CDNA5
//     async-load path (global_load_async_to_lds_b32 + s_wait_asynccnt).
//   * mask/gt (8 MB) fit in the 192 MB L2 -> random 4B gathers are L2 hits.
//   * deterministic 2-stage reductions (block partials in ws, no atomics).
//   * final reduction uses V_WMMA_F32_16X16X4_F32 (exact f32 row-sum vs ones).
// ---------------------------------------------------------------------------

typedef __attribute__((ext_vector_type(2))) float v2f;
typedef __attribute__((ext_vector_type(8))) float v8f;

#define TPB          256      // 8 waves (wave32)
#define WPB          (TPB/32)
#define NODE_BLOCKS  1024
#define E1_BLOCKS    2048
#define E2_BLOCKS    1024
#define WS_STRIDE    2048     // floats per partial-sum array in d_ws

// ---------------- wave / block reductions (wave32) -------------------------

__device__ __forceinline__ float waveReduceSum(float v) {
#pragma unroll
  for (int o = 16; o >= 1; o >>= 1) v += __shfl_xor(v, o, 32);
  return v;
}

// Each call needs its own 8-float LDS slot. Result valid on threadIdx.x==0.
__device__ __forceinline__ float blockReduceSum(float v, float* slot) {
  v = waveReduceSum(v);
  const int lane = threadIdx.x & 31;
  const int w    = threadIdx.x >> 5;
  if (lane == 0) slot[w] = v;
  __syncthreads();
  float r = 0.0f;
  if (w == 0) {
    r = (lane < WPB) ? slot[lane] : 0.0f;
#pragma unroll
    for (int o = 4; o >= 1; o >>= 1) r += __shfl_xor(r, o, 32);
  }
  return r;
}

// ---------------- CDNA5 async global->LDS helpers --------------------------
// VDST vgpr holds the LDS byte offset (low 32 bits of the generic shared
// pointer == offset within the workgroup LDS allocation); VADDR is the 64-bit
// global address. Tracked by ASYNCcnt; loads complete in order.

__device__ __forceinline__ void async_ld_b32(unsigned lds_off, const float* g) {
  unsigned long long ga = (unsigned long long)g;
  asm volatile("global_load_async_to_lds_b32 %0, %1, off"
               :: "v"(lds_off), "v"(ga) : "memory");
}
__device__ __forceinline__ void wait_async_le6() {
  asm volatile("s_wait_asynccnt 6" ::: "memory");
}
__device__ __forceinline__ void wait_async_0() {
  asm volatile("s_wait_asynccnt 0" ::: "memory");
}

// ---------------- kernel 1: node NLL ---------------------------------------

__global__ void node_loss_kernel(const float* __restrict__ pn,
                                 const int* __restrict__ gt,
                                 const int* __restrict__ mask,
                                 int N, int Cn,
                                 float* __restrict__ wsNll,
                                 float* __restrict__ wsCnt) {
  __shared__ float s0[WPB], s1[WPB];
  float nll = 0.0f, cnt = 0.0f;
  for (long long i = (long long)blockIdx.x * TPB + threadIdx.x; i < N;
       i += (long long)NODE_BLOCKS * TPB) {
    if (mask[i] != 0) {
      float p = pn[i * (long long)Cn + gt[i]];
      nll -= logf(p);
      cnt += 1.0f;
    }
  }
  float a = blockReduceSum(nll, s0);
  float b = blockReduceSum(cnt, s1);
  if (threadIdx.x == 0) { wsNll[blockIdx.x] = a; wsCnt[blockIdx.x] = b; }
}

// ---------------- kernel 2: edge squared-diff (async-LDS pipelined) --------
// One wave per edge per iteration: lanes cover row elements {l, l+32} plus a
// tail (Ce-64 elements). Double-buffered: prefetch edge e+step into stage
// st^1 (6 async issues), wait asynccnt<=6 so stage st (issued 6 earlier) is
// complete, then consume stage st from LDS.

__global__ void edge_sqdiff_kernel(const float* __restrict__ pe,
                                   const int* __restrict__ edges,
                                   long long E, int Ce,
                                   float* __restrict__ wsOut) {
  __shared__ float sbuf[WPB][2][2][72];
  __shared__ float rslot[WPB];
  const int lane = threadIdx.x & 31;
  const int w    = threadIdx.x >> 5;
  const int tail = Ce - 64;                       // == 1 for this problem
  const long long step = (long long)E1_BLOCKS * WPB;
  long long e = (long long)blockIdx.x * WPB + w;  // wave-uniform edge index

  auto prefetch = [&](long long eidx, int st) {
    const long long n0 = (long long)edges[2 * eidx];
    const long long n1 = (long long)edges[2 * eidx + 1];
    const float* r0 = pe + n0 * (long long)Ce;
    const float* r1 = pe + n1 * (long long)Ce;
    const unsigned d0 = (unsigned)(unsigned long long)(&sbuf[w][st][0][0]);
    const unsigned d1 = (unsigned)(unsigned long long)(&sbuf[w][st][1][0]);
    async_ld_b32(d0 + 4u * lane,        r0 + lane);        // elems 0..31
    async_ld_b32(d0 + 4u * (32 + lane), r0 + 32 + lane);   // elems 32..63
    async_ld_b32(d1 + 4u * lane,        r1 + lane);
    async_ld_b32(d1 + 4u * (32 + lane), r1 + 32 + lane);
    if (lane < tail) {                                     // elem 64..Ce-1
      async_ld_b32(d0 + 4u * (64 + lane), r0 + 64 + lane);
      async_ld_b32(d1 + 4u * (64 + lane), r1 + 64 + lane);
    }
  };

  float acc = 0.0f;
  int st = 0;
  if (e < E) prefetch(e, 0);
  while (e < E) {
    const long long en = e + step;
    if (en < E) {
      prefetch(en, st ^ 1);   // 6 new in flight
      wait_async_le6();       // in-order completion => stage st is ready
    } else {
      wait_async_0();
    }
    const float a0 = sbuf[w][st][0][lane],      b0 = sbuf[w][st][1][lane];
    const float a1 = sbuf[w][st][0][32 + lane], b1 = sbuf[w][st][1][32 + lane];
    const float d0v = a0 - b0, d1v = a1 - b1;
    acc += d0v * d0v + d1v * d1v;
    if (lane < tail) {
      const float d2 = sbuf[w][st][0][64 + lane] - sbuf[w][st][1][64 + lane];
      acc += d2 * d2;
    }
    e = en;
    st ^= 1;
  }
  const float tot = blockReduceSum(acc, rslot);
  if (threadIdx.x == 0) wsOut[blockIdx.x] = tot;
}

// ---------------- kernel 3: edge-loss terms, intrust, S --------------------
// Thread per edge (grid-stride). mask/gt gathers hit L2 (8 MB arrays).

__global__ void edge_loss_kernel(const float* __restrict__ pe,
                                 const int* __restrict__ gt,
                                 const int* __restrict__ mask,
                                 const int* __restrict__ edges,
                                 long long E, int Ce,
                                 float* __restrict__ wsIntr,
                                 float* __restrict__ wsRaw,
                                 float* __restrict__ wsS) {
  __shared__ float s0[WPB], s1[WPB], s2[WPB];
  float intr = 0.0f, raw = 0.0f, S = 0.0f;
  for (long long e = (long long)blockIdx.x * TPB + threadIdx.x; e < E;
       e += (long long)E2_BLOCKS * TPB) {
    const int n0 = edges[2 * e];
    const int n1 = edges[2 * e + 1];
    const long long rowbase = e * (long long)Ce;
    const float plast = pe[rowbase + (Ce - 1)];
    intr += 1.0f - plast;
    const int m0 = (mask[n0] != 0);
    const int m1 = (mask[n1] != 0);
    if (m0 | m1) {
      S += 1.0f;
      float t;
      if (m0 & m1) {
        const int g0 = gt[n0], g1 = gt[n1];
        t = (g0 == g1) ? pe[rowbase + g0] : plast;
      } else if (m0) {
        t = plast + pe[rowbase + gt[n0]];
      } else {
        t = plast + pe[rowbase + gt[n1]];
      }
      raw -= logf(t);
    }
  }
  const float a = blockReduceSum(intr, s0);
  const float b = blockReduceSum(raw,  s1);
  const float c = blockReduceSum(S,    s2);
  if (threadIdx.x == 0) {
    wsIntr[blockIdx.x] = a; wsRaw[blockIdx.x] = b; wsS[blockIdx.x] = c;
  }
}

// ---------------- kernel 4: final combine (WMMA f32 reduction) -------------
// A (16x4 f32, 2 VGPRs across 32 lanes = 64 partials) x B = ones gives exact
// f32 row sums in D; 8 VGPR adds + shfl_xor(16) complete the reduction.

__device__ __forceinline__ float waveWmmaSum(const float* __restrict__ arr, int n) {
  const int lane = threadIdx.x & 31;
  float a0 = 0.0f, a1 = 0.0f;
  for (int base = 0; base < n; base += 64) {
    const int i0 = base + lane, i1 = base + 32 + lane;
    if (i0 < n) a0 += arr[i0];
    if (i1 < n) a1 += arr[i1];
  }
  v2f A; A.x = a0; A.y = a1;          // A[m][{0,1}] lanes 0-15, A[m][{2,3}] lanes 16-31
  v2f B; B.x = 1.0f; B.y = 1.0f;      // all-ones 4x16 regardless of striping
  v8f C = {};
  v8f D = __builtin_amdgcn_wmma_f32_16x16x4_f32(false, A, false, B,
                                                (short)0, C, false, false);
  float s = D[0] + D[1] + D[2] + D[3] + D[4] + D[5] + D[6] + D[7];
  s += __shfl_xor(s, 16, 32);         // rows 0-7 sum + rows 8-15 sum
  return s;
}

__global__ void final_kernel(const float* __restrict__ ws, float* __restrict__ out) {
  const float A0 = waveWmmaSum(ws + 0 * WS_STRIDE, NODE_BLOCKS); // sum -log p (masked)
  const float A1 = waveWmmaSum(ws + 1 * WS_STRIDE, NODE_BLOCKS); // sum mask
  const float A2 = waveWmmaSum(ws + 2 * WS_STRIDE, E1_BLOCKS);   // sqdiff
  const float A3 = waveWmmaSum(ws + 3 * WS_STRIDE, E2_BLOCKS);   // sum(1-intrust)
  const float A4 = waveWmmaSum(ws + 4 * WS_STRIDE, E2_BLOCKS);   // raw edge loss
  const float A5 = waveWmmaSum(ws + 5 * WS_STRIDE, E2_BLOCKS);   // S
  if (threadIdx.x == 0) {
    const float loss = A0 / A1;
    const float semi = 0.5f * A3 * A2;                 // SEMI_LAMBDA = 0.5
    const float edge = A4 * (1.0f * 1.0f) / (A5 * A5); // EDGE_LAMBDA = 1.0
    out[0] = loss + semi + edge;
  }
}

// ---------------- launch ---------------------------------------------------

extern "C" void kernel_launch(void* const* d_in, const int* in_sizes, int n_in,
                              void* d_out, int out_size, void* d_ws, size_t ws_size,
                              hipStream_t stream) {
  const float* pn    = (const float*)d_in[0];   // poss_node (N, Cn)
  const float* pe    = (const float*)d_in[1];   // poss_edge (E, Ce)
  const int*   gt    = (const int*)d_in[2];     // groundTruth (N,)
  const int*   mask  = (const int*)d_in[3];     // mask (N,) 0/1
  const int*   edges = (const int*)d_in[4];     // edges (E, 2)

  const int       N  = in_sizes[2];
  const long long E  = (long long)in_sizes[4] / 2;
  const int       Cn = in_sizes[0] / N;               // 64
  const int       Ce = (int)(in_sizes[1] / E);        // 65

  float* ws = (float*)d_ws;  // 6 arrays of WS_STRIDE floats, fully overwritten

  node_loss_kernel<<<NODE_BLOCKS, TPB, 0, stream>>>(
      pn, gt, mask, N, Cn, ws + 0 * WS_STRIDE, ws + 1 * WS_STRIDE);
  edge_sqdiff_kernel<<<E1_BLOCKS, TPB, 0, stream>>>(
      pe, edges, E, Ce, ws + 2 * WS_STRIDE);
  edge_loss_kernel<<<E2_BLOCKS, TPB, 0, stream>>>(
      pe, gt, mask, edges, E, Ce,
      ws + 3 * WS_STRIDE, ws + 4 * WS_STRIDE, ws + 5 * WS_STRIDE);
  final_kernel<<<1, 32, 0, stream>>>(ws, (float*)d_out);
}